// IMUPreintegrator_29858612642446
// MI455X (gfx1250) — compile-verified
//
#include <hip/hip_runtime.h>
#include <hip/hip_bf16.h>

typedef __attribute__((ext_vector_type(2))) float v2f;
typedef __attribute__((ext_vector_type(8))) float v8f;

#define F_LEN 2048
#define LDS_STRIDE 17                       // 16x16 tile padded -> conflict-free transposed reads
#define GYRO_COV (0.00016968f * 0.00016968f)
#define ACC_COV  (0.002f * 0.002f)
#define GRAVITY  9.81007f

__device__ __forceinline__ float sel3(int i, float a, float b, float c) {
    return (i == 0) ? a : ((i == 1) ? b : c);
}

// Wave-local LDS ordering: one wave owns its tiles, lanes run in lockstep, so a
// dscnt drain + compiler barrier is sufficient (no s_barrier needed).
__device__ __forceinline__ void wave_lds_fence() {
    __builtin_amdgcn_wave_barrier();
    asm volatile("s_wait_dscnt 0" ::: "memory");
    __builtin_amdgcn_wave_barrier();
}

__global__ __launch_bounds__(256, 1)
void imu_preintegrate_wmma(const float* __restrict__ dtp,
                           const float* __restrict__ gyro,
                           const float* __restrict__ acc,
                           const float* __restrict__ init_pos,
                           const float* __restrict__ init_rot,
                           const float* __restrict__ init_vel,
                           float* __restrict__ out, int B) {
    __shared__ float smem[8][2][16 * LDS_STRIDE];

    const int wave = threadIdx.x >> 5;
    const int lane = threadIdx.x & 31;
    const int b = blockIdx.x * 8 + wave;
    if (b >= B) return;                       // uniform per wave -> EXEC stays all-1 for WMMA

    float* covL = smem[wave][0];
    float* PL   = smem[wave][1];

    const int n = lane & 15;                  // B/C/D column owned by this lane; also A row m
    const int h = lane >> 4;                  // lane half (K/M offset selector)

    // zero-init 16x16 covariance tile (padding rows/cols stay zero forever)
#pragma unroll
    for (int i = 0; i < 9; ++i) {
        int idx = lane + 32 * i;
        if (idx < 16 * LDS_STRIDE) covL[idx] = 0.f;
    }
    wave_lds_fence();

    // initial state (uniform, tiny)
    const float r0_00 = init_rot[0], r0_01 = init_rot[1], r0_02 = init_rot[2];
    const float r0_10 = init_rot[3], r0_11 = init_rot[4], r0_12 = init_rot[5];
    const float r0_20 = init_rot[6], r0_21 = init_rot[7], r0_22 = init_rot[8];
    const float p0x = init_pos[0], p0y = init_pos[1], p0z = init_pos[2];
    const float v0x = init_vel[0], v0y = init_vel[1], v0z = init_vel[2];

    // accumulated delta state (replicated per lane)
    float D00 = 1.f, D01 = 0.f, D02 = 0.f;
    float D10 = 0.f, D11 = 1.f, D12 = 0.f;
    float D20 = 0.f, D21 = 0.f, D22 = 1.f;
    float Dvx = 0.f, Dvy = 0.f, Dvz = 0.f;
    float Dpx = 0.f, Dpy = 0.f, Dpz = 0.f;
    float tacc = 0.f;

    const size_t BF = (size_t)B * F_LEN;
    const size_t VEL_OFF = BF * 9;
    const size_t POS_OFF = VEL_OFF + BF * 3;
    const size_t COV_OFF = POS_OFF + BF * 3;
    const size_t baseBF = (size_t)b * F_LEN;

    for (int f = 0; f < F_LEN; ++f) {
        const size_t sidx = baseBF + f;
        const float dtk = __builtin_nontemporal_load(&dtp[sidx]);
        const float wx = __builtin_nontemporal_load(&gyro[sidx * 3 + 0]);
        const float wy = __builtin_nontemporal_load(&gyro[sidx * 3 + 1]);
        const float wz = __builtin_nontemporal_load(&gyro[sidx * 3 + 2]);
        const float ax = __builtin_nontemporal_load(&acc[sidx * 3 + 0]);
        const float ay = __builtin_nontemporal_load(&acc[sidx * 3 + 1]);
        const float az = __builtin_nontemporal_load(&acc[sidx * 3 + 2]);

        // ---- SO3 Exp map + right Jacobian of w*dt ----
        const float wvx = wx * dtk, wvy = wy * dtk, wvz = wz * dtk;
        const float xx = wvx * wvx, yy = wvy * wvy, zz = wvz * wvz;
        const float xy = wvx * wvy, xz = wvx * wvz, yz = wvy * wvz;
        const float t2 = xx + yy + zz;
        const bool  sm = t2 < 1e-8f;
        const float t2s = sm ? 1.f : t2;
        const float tt = sqrtf(t2s);
        const float sn = __sinf(tt), cs = __cosf(tt);
        const float cA = sm ? 1.f - t2 * (1.f / 6.f)  : sn / tt;
        const float cB = sm ? 0.5f - t2 * (1.f / 24.f) : (1.f - cs) / t2s;
        const float cC = sm ? (1.f / 6.f) - t2 * (1.f / 120.f) : (tt - sn) / (t2s * tt);

        const float dr00 = 1.f - cB * (yy + zz), dr01 = -cA * wvz + cB * xy, dr02 =  cA * wvy + cB * xz;
        const float dr10 =  cA * wvz + cB * xy,  dr11 = 1.f - cB * (xx + zz), dr12 = -cA * wvx + cB * yz;
        const float dr20 = -cA * wvy + cB * xz,  dr21 =  cA * wvx + cB * yz,  dr22 = 1.f - cB * (xx + yy);

        const float j00 = 1.f - cC * (yy + zz), j01 =  cB * wvz + cC * xy, j02 = -cB * wvy + cC * xz;
        const float j10 = -cB * wvz + cC * xy,  j11 = 1.f - cC * (xx + zz), j12 =  cB * wvx + cC * yz;
        const float j20 =  cB * wvy + cC * xz,  j21 = -cB * wvx + cC * yz,  j22 = 1.f - cC * (xx + yy);

        // Ha = Dr(old) @ skew(a)
        const float ha00 = D01 * az - D02 * ay, ha01 = D02 * ax - D00 * az, ha02 = D00 * ay - D01 * ax;
        const float ha10 = D11 * az - D12 * ay, ha11 = D12 * ax - D10 * az, ha12 = D10 * ay - D11 * ax;
        const float ha20 = D21 * az - D22 * ay, ha21 = D22 * ax - D20 * az, ha22 = D20 * ay - D21 * ax;

        // ---- row m of the 9x9 propagation matrix A (m = lane%16; rows >=9 are zero) ----
        const int m = n;
        float ar[16];
#pragma unroll
        for (int k = 0; k < 16; ++k) ar[k] = 0.f;
        {
            const float s1 = -dtk, s2 = -0.5f * dtk * dtk;
            const int mr = (m < 3) ? m : ((m < 6) ? m - 3 : m - 6);
            const float top0 = sel3(mr, dr00, dr01, dr02);   // dR^T[m][0] = dR[0][m]
            const float top1 = sel3(mr, dr10, dr11, dr12);
            const float top2 = sel3(mr, dr20, dr21, dr22);
            const float hc0 = sel3(mr, ha00, ha10, ha20);    // Ha[mr][k]
            const float hc1 = sel3(mr, ha01, ha11, ha21);
            const float hc2 = sel3(mr, ha02, ha12, ha22);
            ar[0] = (m < 3) ? top0 : ((m < 6) ? s1 * hc0 : ((m < 9) ? s2 * hc0 : 0.f));
            ar[1] = (m < 3) ? top1 : ((m < 6) ? s1 * hc1 : ((m < 9) ? s2 * hc1 : 0.f));
            ar[2] = (m < 3) ? top2 : ((m < 6) ? s1 * hc2 : ((m < 9) ? s2 * hc2 : 0.f));
            ar[3] = (m == 3) ? 1.f : ((m == 6) ? dtk : 0.f);
            ar[4] = (m == 4) ? 1.f : ((m == 7) ? dtk : 0.f);
            ar[5] = (m == 5) ? 1.f : ((m == 8) ? dtk : 0.f);
            ar[6] = (m == 6) ? 1.f : 0.f;
            ar[7] = (m == 7) ? 1.f : 0.f;
            ar[8] = (m == 8) ? 1.f : 0.f;
        }
        // A-operand fragment: reg j holds A[m][k0(j)+2h]
        float af[8];
#pragma unroll
        for (int j = 0; j < 8; ++j) {
            const int k0 = 4 * (j >> 1) + (j & 1);
            af[j] = h ? ar[k0 + 2] : ar[k0];
        }

        // ---- noise matrix N, column n (symmetric; Dr Dr^T == I exploited) ----
        const float gscale = GYRO_COV * dtk * dtk;
        const float na2 = ACC_COV * dtk * dtk;
        const float na3 = 0.5f * ACC_COV * dtk * dtk * dtk;
        const float na4 = 0.25f * ACC_COV * dtk * dtk * dtk * dtk;
        const float g00 = j00 * j00 + j01 * j01 + j02 * j02;
        const float g01 = j00 * j10 + j01 * j11 + j02 * j12;
        const float g02 = j00 * j20 + j01 * j21 + j02 * j22;
        const float g11 = j10 * j10 + j11 * j11 + j12 * j12;
        const float g12 = j10 * j20 + j11 * j21 + j12 * j22;
        const float g22 = j20 * j20 + j21 * j21 + j22 * j22;
        const float nc0 = (n < 3) ? gscale * sel3(n, g00, g01, g02) : 0.f;
        const float nc1 = (n < 3) ? gscale * sel3(n, g01, g11, g12) : 0.f;
        const float nc2 = (n < 3) ? gscale * sel3(n, g02, g12, g22) : 0.f;
        const float nc3 = (n == 3) ? na2 : ((n == 6) ? na3 : 0.f);
        const float nc4 = (n == 4) ? na2 : ((n == 7) ? na3 : 0.f);
        const float nc5 = (n == 5) ? na2 : ((n == 8) ? na3 : 0.f);
        const float nc6 = (n == 6) ? na4 : ((n == 3) ? na3 : 0.f);
        const float nc7 = (n == 7) ? na4 : ((n == 4) ? na3 : 0.f);
        const float nc8 = (n == 8) ? na4 : ((n == 5) ? na3 : 0.f);

        // ---- P = A @ cov  (cov as B operand from LDS; K-chunks 3 is all-zero -> skipped) ----
        float bc[8];
#pragma unroll
        for (int j = 0; j < 8; ++j) {
            const int k0 = 4 * (j >> 1) + (j & 1);
            bc[j] = covL[(k0 + 2 * h) * LDS_STRIDE + n];
        }
        v8f accP = {};
#pragma unroll
        for (int c = 0; c < 3; ++c) {
            v2f av; av[0] = af[2 * c]; av[1] = af[2 * c + 1];
            v2f bv; bv[0] = bc[2 * c]; bv[1] = bc[2 * c + 1];
            accP = __builtin_amdgcn_wmma_f32_16x16x4_f32(false, av, false, bv,
                                                         (short)0, accP, false, false);
        }
#pragma unroll
        for (int r = 0; r < 8; ++r) PL[(r + 8 * h) * LDS_STRIDE + n] = accP[r];
        wave_lds_fence();

        // ---- cov' = A @ P^T + N   (== A cov A^T + N since cov symmetric) ----
        float bp[8];
#pragma unroll
        for (int j = 0; j < 8; ++j) {
            const int k0 = 4 * (j >> 1) + (j & 1);
            bp[j] = PL[n * LDS_STRIDE + k0 + 2 * h];    // stride-17 -> conflict-free
        }
        v8f accC;
        accC[0] = h ? nc8 : nc0;
        accC[1] = h ? 0.f : nc1;
        accC[2] = h ? 0.f : nc2;
        accC[3] = h ? 0.f : nc3;
        accC[4] = h ? 0.f : nc4;
        accC[5] = h ? 0.f : nc5;
        accC[6] = h ? 0.f : nc6;
        accC[7] = h ? 0.f : nc7;
#pragma unroll
        for (int c = 0; c < 3; ++c) {
            v2f av; av[0] = af[2 * c]; av[1] = af[2 * c + 1];
            v2f bv; bv[0] = bp[2 * c]; bv[1] = bp[2 * c + 1];
            accC = __builtin_amdgcn_wmma_f32_16x16x4_f32(false, av, false, bv,
                                                         (short)0, accC, false, false);
        }
#pragma unroll
        for (int r = 0; r < 8; ++r) covL[(r + 8 * h) * LDS_STRIDE + n] = accC[r];
        wave_lds_fence();

        // ---- state update ----
        const float Rax = D00 * ax + D01 * ay + D02 * az;
        const float Ray = D10 * ax + D11 * ay + D12 * az;
        const float Raz = D20 * ax + D21 * ay + D22 * az;
        const float hdt2 = 0.5f * dtk * dtk;
        Dpx += Dvx * dtk + Rax * hdt2;
        Dpy += Dvy * dtk + Ray * hdt2;
        Dpz += Dvz * dtk + Raz * hdt2;
        Dvx += Rax * dtk; Dvy += Ray * dtk; Dvz += Raz * dtk;
        const float nD00 = D00 * dr00 + D01 * dr10 + D02 * dr20;
        const float nD01 = D00 * dr01 + D01 * dr11 + D02 * dr21;
        const float nD02 = D00 * dr02 + D01 * dr12 + D02 * dr22;
        const float nD10 = D10 * dr00 + D11 * dr10 + D12 * dr20;
        const float nD11 = D10 * dr01 + D11 * dr11 + D12 * dr21;
        const float nD12 = D10 * dr02 + D11 * dr12 + D12 * dr22;
        const float nD20 = D20 * dr00 + D21 * dr10 + D22 * dr20;
        const float nD21 = D20 * dr01 + D21 * dr11 + D22 * dr21;
        const float nD22 = D20 * dr02 + D21 * dr12 + D22 * dr22;
        D00 = nD00; D01 = nD01; D02 = nD02;
        D10 = nD10; D11 = nD11; D12 = nD12;
        D20 = nD20; D21 = nD21; D22 = nD22;
        tacc += dtk;

        // ---- lane-distributed output store: lanes 0-8 rot, 9-11 vel, 12-14 pos ----
        float outval = 0.f;
        size_t off = 0;
        if (lane < 9) {
            const int i = lane / 3, j = lane % 3;
            const float a0 = sel3(i, r0_00, r0_10, r0_20);
            const float a1 = sel3(i, r0_01, r0_11, r0_21);
            const float a2 = sel3(i, r0_02, r0_12, r0_22);
            const float b0 = sel3(j, D00, D01, D02);
            const float b1 = sel3(j, D10, D11, D12);
            const float b2 = sel3(j, D20, D21, D22);
            outval = a0 * b0 + a1 * b1 + a2 * b2;
            off = sidx * 9 + lane;
        } else if (lane < 12) {
            const int c = lane - 9;
            const float a0 = sel3(c, r0_00, r0_10, r0_20);
            const float a1 = sel3(c, r0_01, r0_11, r0_21);
            const float a2 = sel3(c, r0_02, r0_12, r0_22);
            const float g = (c == 2) ? GRAVITY : 0.f;
            outval = sel3(c, v0x, v0y, v0z) + (a0 * Dvx + a1 * Dvy + a2 * Dvz) - g * tacc;
            off = VEL_OFF + sidx * 3 + c;
        } else if (lane < 15) {
            const int c = lane - 12;
            const float a0 = sel3(c, r0_00, r0_10, r0_20);
            const float a1 = sel3(c, r0_01, r0_11, r0_21);
            const float a2 = sel3(c, r0_02, r0_12, r0_22);
            const float g = (c == 2) ? GRAVITY : 0.f;
            outval = sel3(c, p0x, p0y, p0z) + sel3(c, v0x, v0y, v0z) * tacc
                   + (a0 * Dpx + a1 * Dpy + a2 * Dpz) - 0.5f * g * tacc * tacc;
            off = POS_OFF + sidx * 3 + c;
        }
        if (lane < 15) __builtin_nontemporal_store(outval, &out[off]);
    }

    // final 9x9 covariance
    for (int idx = lane; idx < 81; idx += 32) {
        out[COV_OFF + (size_t)b * 81 + idx] = covL[(idx / 9) * LDS_STRIDE + (idx % 9)];
    }
}

extern "C" void kernel_launch(void* const* d_in, const int* in_sizes, int n_in,
                              void* d_out, int out_size, void* d_ws, size_t ws_size,
                              hipStream_t stream) {
    const float* dt       = (const float*)d_in[0];
    const float* gyro     = (const float*)d_in[1];
    const float* acc      = (const float*)d_in[2];
    const float* init_pos = (const float*)d_in[3];
    const float* init_rot = (const float*)d_in[4];
    const float* init_vel = (const float*)d_in[5];
    float* out = (float*)d_out;

    const int B = in_sizes[0] / F_LEN;           // dt is (B, F, 1)
    dim3 block(256);                              // 8 wave32 per block, one sequence per wave
    dim3 grid((B + 7) / 8);
    imu_preintegrate_wmma<<<grid, block, 0, stream>>>(dt, gyro, acc, init_pos,
                                                      init_rot, init_vel, out, B);
}